// TestingModel1_5239860101887
// MI455X (gfx1250) — compile-verified
//
#include <hip/hip_runtime.h>
#include <stdint.h>

// GraphAttentionPooling, p=3, F=512. Groups are 1536 contiguous f32.
// Memory-bound (~0.5 flop/byte): use the CDNA5 Tensor Data Mover to stage
// each group into LDS (per-wave DMA, TENSORcnt), compute from LDS/registers.

#define P 3
#define FDIM 512
#define GROUP_ELEMS (P * FDIM)                 // 1536 floats = 6 KB
#define GROUPS_PER_BLOCK 8                      // one group per wave32
#define TILE_ELEMS (GROUPS_PER_BLOCK * GROUP_ELEMS)

typedef unsigned int u32x4 __attribute__((ext_vector_type(4)));
typedef int          i32x4 __attribute__((ext_vector_type(4)));
typedef int          i32x8 __attribute__((ext_vector_type(8)));

// Generic pointer -> 32-bit LDS byte offset (addrspacecast flat->local).
__device__ __forceinline__ uint32_t lds_byte_offset(const void* p) {
  return (uint32_t)(uintptr_t)(__attribute__((address_space(3))) const void*)p;
}

__global__ __launch_bounds__(256)
void gap_pool_tdm_kernel(const float* __restrict__ x,
                         const float* __restrict__ Ww,
                         const float* __restrict__ Wb,
                         float* __restrict__ out,
                         int n_groups) {
  __shared__ __align__(16) float smem[TILE_ELEMS];

  const int lane = (int)(threadIdx.x & 31);
  // Wave-uniform values, forced scalar for the TDM descriptor.
  const int wave = __builtin_amdgcn_readfirstlane((int)(threadIdx.x >> 5));
  const int g    = __builtin_amdgcn_readfirstlane(
                       (int)(blockIdx.x * GROUPS_PER_BLOCK) + wave);
  if (g >= n_groups) return;  // wave-uniform branch

  // ---------------- TDM: DMA this wave's group (6 KB) global -> LDS --------
  const uint32_t lds_addr =
      lds_byte_offset(&smem[0]) + (uint32_t)wave * (GROUP_ELEMS * 4u);
  const uint64_t gaddr = (uint64_t)(uintptr_t)(x + (size_t)g * GROUP_ELEMS);

  // D# group 0: count=1 (valid user descriptor), lds_addr, global_addr, type=2
  u32x4 d0;
  d0.x = 1u;                                        // count=1, is_restore=0
  d0.y = lds_addr;                                  // lds_addr [63:32]
  d0.z = (uint32_t)(gaddr & 0xffffffffu);           // global_addr [95:64]
  d0.w = (uint32_t)((gaddr >> 32) & 0x01ffffffu)    // global_addr [120:96]
       | (2u << 30);                                // type=2 ("image")

  // D# group 1: data_size=4B, 1-D tile of 1536 elems, no multicast/pad/iter
  i32x8 d1;
  d1[0] = (int)(2u << 16);                          // data_size=2 (4 bytes)
  d1[1] = (int)((uint32_t)GROUP_ELEMS << 16);       // tensor_dim0[15:0]
  d1[2] = (int)(1u << 16);                          // tensor_dim0 hi=0, tensor_dim1=1
  d1[3] = (int)((uint32_t)GROUP_ELEMS << 16);       // tile_dim0 = 1536
  d1[4] = 1;                                        // tile_dim1 = 1, tile_dim2 = 0
  d1[5] = GROUP_ELEMS;                              // tensor_dim0_stride lo32
  d1[6] = (int)((uint32_t)GROUP_ELEMS << 16);       // stride hi=0, dim1_stride lo16
  d1[7] = 0;

  i32x4 dz4 = {0, 0, 0, 0};                         // groups 2/3 unused (<=2D)
  i32x8 dz8 = {0, 0, 0, 0, 0, 0, 0, 0};             // extra arg on 6-arg builtin

  __builtin_amdgcn_tensor_load_to_lds(d0, d1, dz4, dz4, dz8, 0);
  __builtin_amdgcn_s_wait_tensorcnt(0);             // this wave's DMA done
  asm volatile("" ::: "memory");                    // keep ds reads after wait

  // ---------------- Compute entirely within this wave ----------------------
  const float* s = &smem[wave * GROUP_ELEMS];

  // Lane l owns columns f = c*128 + l*4 .. +3, c=0..3 (16 cols, float4 loads).
  float4 xr[P][4];
  float p0 = 0.f, p1 = 0.f, p2 = 0.f;
#pragma unroll
  for (int c = 0; c < 4; ++c) {
    const int f = c * 128 + lane * 4;
    const float4 wc = *(const float4*)(Ww + f);
#pragma unroll
    for (int j = 0; j < P; ++j) {
      const float4 v = *(const float4*)(s + j * FDIM + f);
      xr[j][c] = v;
      const float d = v.x * wc.x + v.y * wc.y + v.z * wc.z + v.w * wc.w;
      if (j == 0) p0 += d; else if (j == 1) p1 += d; else p2 += d;
    }
  }

  // Wave32 reduction of the three partial dot products.
#pragma unroll
  for (int off = 16; off > 0; off >>= 1) {
    p0 += __shfl_xor(p0, off, 32);
    p1 += __shfl_xor(p1, off, 32);
    p2 += __shfl_xor(p2, off, 32);
  }

  // 3-way softmax (bias cancels in softmax but matches the reference).
  const float b  = Wb[0];
  const float l0 = p0 + b, l1 = p1 + b, l2 = p2 + b;
  const float m  = fmaxf(l0, fmaxf(l1, l2));
  const float e0 = __expf(l0 - m), e1 = __expf(l1 - m), e2 = __expf(l2 - m);
  const float inv = 1.0f / (e0 + e1 + e2);
  const float a0 = e0 * inv, a1 = e1 * inv, a2 = e2 * inv;

  // out[g*512 + f] = a0*x0[f] + a1*x1[f] + a2*x2[f]  (coalesced float4 stores)
  float* o = out + (size_t)g * FDIM;
#pragma unroll
  for (int c = 0; c < 4; ++c) {
    const int f = c * 128 + lane * 4;
    float4 r;
    r.x = a0 * xr[0][c].x + a1 * xr[1][c].x + a2 * xr[2][c].x;
    r.y = a0 * xr[0][c].y + a1 * xr[1][c].y + a2 * xr[2][c].y;
    r.z = a0 * xr[0][c].z + a1 * xr[1][c].z + a2 * xr[2][c].z;
    r.w = a0 * xr[0][c].w + a1 * xr[1][c].w + a2 * xr[2][c].w;
    *(float4*)(o + f) = r;
  }
}

extern "C" void kernel_launch(void* const* d_in, const int* in_sizes, int n_in,
                              void* d_out, int out_size, void* d_ws, size_t ws_size,
                              hipStream_t stream) {
  const float* x  = (const float*)d_in[0];   // batch_tensor [B, N, F] f32
  const float* Ww = (const float*)d_in[1];   // [F, 1] f32
  const float* Wb = (const float*)d_in[2];   // [1] f32
  float* out = (float*)d_out;                // [B, steps, F, 1] f32

  const int total_in = in_sizes[0];
  const int n_groups = total_in / GROUP_ELEMS;         // 16384 for ref shapes
  const int blocks   = (n_groups + GROUPS_PER_BLOCK - 1) / GROUPS_PER_BLOCK;

  hipLaunchKernelGGL(gap_pool_tdm_kernel, dim3(blocks), dim3(256), 0, stream,
                     x, Ww, Wb, out, n_groups);
}